// ModelNew_73315091744391
// MI455X (gfx1250) — compile-verified
//
#include <hip/hip_runtime.h>

// Row-wise inclusive prefix product (cumprod along dim=1).
// Rows: 1024, row length: 32768, f32 in / f32 out.
// Memory-bound: 256 MB total traffic -> ~11 us floor at 23.3 TB/s.
// Strategy: 1 workgroup per row, double-buffered async global->LDS staging
// (ASYNCcnt path), wave32 shuffle scan, non-temporal streaming stores.

#define ROWLEN  32768
#define THREADS 256
#define WAVES   (THREADS / 32)
#define SEG     32                    // contiguous floats per thread per chunk
#define CHUNK   (THREADS * SEG)       // 8192 floats = 32 KB
#define NCHUNK  (ROWLEN / CHUNK)      // 4

typedef __attribute__((ext_vector_type(4))) float v4f;

// Async DMA of 16 bytes from global memory into LDS (gfx1250 ASYNC path).
// VDST = VGPR holding LDS byte address, VADDR = 64-bit global address (GV mode).
__device__ __forceinline__ void async_load16(unsigned lds_byte, const float* gsrc) {
  asm volatile("global_load_async_to_lds_b128 %0, %1, off"
               :
               : "v"(lds_byte), "v"(gsrc)
               : "memory");
}

__device__ __forceinline__ void wait_async_zero() {
#if __has_builtin(__builtin_amdgcn_s_wait_asynccnt)
  __builtin_amdgcn_s_wait_asynccnt(0);
#else
  asm volatile("s_wait_asynccnt 0" ::: "memory");
#endif
}

__global__ void __launch_bounds__(THREADS)
cumprod_rows_kernel(const float* __restrict__ x, float* __restrict__ y) {
  __shared__ __align__(16) float buf[2 * CHUNK];   // 64 KB double buffer
  __shared__ float s_wtot[WAVES];
  __shared__ float s_wpre[WAVES + 1];              // [WAVES] = chunk total

  const int tid       = threadIdx.x;
  const unsigned lane = tid & 31u;
  const unsigned wid  = (unsigned)tid >> 5;

  const float* xr = x + (size_t)blockIdx.x * ROWLEN;
  float*       yr = y + (size_t)blockIdx.x * ROWLEN;

  // Low 32 bits of a flat pointer into LDS == byte offset within the wave's
  // LDS allocation (aperture mapping truncates to addr[31:0]).
  const unsigned ldsBase = (unsigned)(unsigned long long)&buf[0];

  // ---- prologue: async-stage chunk 0 into buffer 0 ----
#pragma unroll
  for (int j = 0; j < SEG / 4; ++j) {
    const int e = (j * THREADS + tid) * 4;         // float index within chunk
    async_load16(ldsBase + (unsigned)(e * 4), xr + e);
  }
  wait_async_zero();
  __syncthreads();

  float R = 1.0f;  // block-uniform running product of all previous chunks

  for (int c = 0; c < NCHUNK; ++c) {
    const int cur = c & 1;

    // ---- prefetch chunk c+1 into the other half (overlaps with compute) ----
    if (c + 1 < NCHUNK) {
      const int nxt = cur ^ 1;
      const float* src = xr + (c + 1) * CHUNK;
#pragma unroll
      for (int j = 0; j < SEG / 4; ++j) {
        const int e = (j * THREADS + tid) * 4;
        async_load16(ldsBase + (unsigned)((nxt * CHUNK + e) * 4), src + e);
      }
    }

    const float* l = &buf[cur * CHUNK + tid * SEG];

    // ---- pass 1: product of this thread's SEG contiguous elements ----
    float p = 1.0f;
#pragma unroll
    for (int j = 0; j < SEG / 4; ++j) {
      v4f t = *(const v4f*)(l + 4 * j);            // ds_load_b128
      p *= t.x * t.y * t.z * t.w;
    }

    // ---- wave32 inclusive scan of per-thread products ----
    float inc = p;
#pragma unroll
    for (int d = 1; d < 32; d <<= 1) {
      float n = __shfl_up(inc, (unsigned)d, 32);
      if (lane >= (unsigned)d) inc *= n;
    }
    if (lane == 31) s_wtot[wid] = inc;
    __syncthreads();
    if (tid == 0) {
      float run = 1.0f;
#pragma unroll
      for (int w = 0; w < WAVES; ++w) { s_wpre[w] = run; run *= s_wtot[w]; }
      s_wpre[WAVES] = run;                          // whole-chunk product
    }
    __syncthreads();

    float exc = __shfl_up(inc, 1u, 32);             // exclusive within wave
    if (lane == 0) exc = 1.0f;

    float run = R * s_wpre[wid] * exc;              // this thread's start value
    const float chunk_total = s_wpre[WAVES];

    // ---- pass 2: apply running prefix, stream out with NT stores ----
    float* op = yr + c * CHUNK + tid * SEG;
#pragma unroll
    for (int j = 0; j < SEG / 4; ++j) {
      v4f t = *(const v4f*)(l + 4 * j);
      v4f o;
      run *= t.x; o.x = run;
      run *= t.y; o.y = run;
      run *= t.z; o.z = run;
      run *= t.w; o.w = run;
      __builtin_nontemporal_store(o, (v4f*)(op + 4 * j));
    }

    R *= chunk_total;

    // next chunk fully staged + all waves done with both buffers before swap
    wait_async_zero();
    __syncthreads();
  }
}

extern "C" void kernel_launch(void* const* d_in, const int* in_sizes, int n_in,
                              void* d_out, int out_size, void* d_ws, size_t ws_size,
                              hipStream_t stream) {
  (void)n_in; (void)out_size; (void)d_ws; (void)ws_size;
  const float* x = (const float*)d_in[0];
  float* y = (float*)d_out;
  const int rows = in_sizes[0] / ROWLEN;            // 1024
  cumprod_rows_kernel<<<dim3(rows), dim3(THREADS), 0, stream>>>(x, y);
}